// ConnectivityLoss_42477226557607
// MI455X (gfx1250) — compile-verified
//
#include <hip/hip_runtime.h>
#include <hip/hip_bf16.h>
#include <cstdint>

// ConnectivityLoss (MALIS) for MI455X / gfx1250.
//
// Mapping rationale (no runtime available -> reason from HW specs):
//  * Inputs are 2 x 1MB; at 23.3 TB/s HBM this is ~0.1us -> memory is a
//    non-issue. The cost is 512 independent, inherently serial MALIS
//    problems (stable sort of 1984 edges + union-find + histogram merges).
//    No GEMM structure -> WMMA is inapplicable; this is latency-bound
//    integer/branch work, so the win is parallelism ACROSS windows:
//    512 single-wave32 workgroups spread over the WGPs.
//  * All random-access state (union-find parents, label histograms as
//    intrusive sorted linked lists, edge ordering) lives in LDS
//    (~38KB/block; 320KB/WGP lets multiple blocks co-reside).
//  * CDNA5 path: each block stages its strided 32x32 f32 tiles with the
//    Tensor Data Mover (tensor_load_to_lds, one instruction per tile,
//    2D descriptor tile=32x32 / stride=256 elems) + s_wait_tensorcnt.
//  * Weights are never materialized: loss contribution and the
//    normalization sum are accumulated during Kruskal, so each task emits
//    a single partial; a deterministic fixed-order reduction produces the
//    scalar output (no float atomics -> replay-deterministic).

#define WINSZ  32
#define NPIX   (WINSZ * WINSZ)        // 1024
#define NEH    (WINSZ * (WINSZ - 1))  // 992 horizontal edges
#define NEDGE  (2 * NEH)              // 1984 edges
#define NTASK  512                    // 4 batch * 64 windows * 2 passes

typedef uint32_t v4u __attribute__((ext_vector_type(4)));
typedef int      v8i __attribute__((ext_vector_type(8)));
typedef int      v4i __attribute__((ext_vector_type(4)));

#ifndef __has_builtin
#define __has_builtin(x) 0
#endif

__device__ __forceinline__ int uf_find(unsigned short* p, int x) {
  while ((int)p[x] != x) {          // path halving
    p[x] = p[p[x]];
    x = (int)p[x];
  }
  return x;
}

// Edge e -> pixel pair, matching the reference edge enumeration:
// first 992 horizontal (row-major over 32x31), then 992 vertical (31x32).
__device__ __forceinline__ void edge_nodes(int e, int& a, int& b) {
  if (e < NEH) {
    int r = e / 31, c = e % 31;
    a = r * WINSZ + c;
    b = a + 1;
  } else {
    int k = e - NEH;      // row-major over (31,32): flat index == pixel index
    a = k;
    b = k + WINSZ;
  }
}

__global__ __launch_bounds__(32)
void malis_window_kernel(const float* __restrict__ y_true,
                         const float* __restrict__ y_pred,
                         float* __restrict__ partials)
{
  __shared__ float          tile_p[NPIX];   // 4 KB
  __shared__ float          tile_t[NPIX];   // 4 KB
  __shared__ float          cost[NEDGE];    // 7.75 KB
  __shared__ unsigned short order[NEDGE];   // 3.9 KB
  __shared__ unsigned short auxb[NEDGE];    // 3.9 KB
  __shared__ unsigned short parent[NPIX];   // 2 KB
  __shared__ unsigned short seg[NPIX];      // 2 KB
  __shared__ unsigned short csize[NPIX];    // 2 KB  labeled-node count per root
  __shared__ short          head[NPIX];     // 2 KB  histogram list head per root
  __shared__ unsigned short hlab[NPIX];     // 2 KB  histogram entry: label
  __shared__ unsigned short hcnt[NPIX];     // 2 KB  histogram entry: count
  __shared__ short          hnext[NPIX];    // 2 KB  histogram entry: next

  const int task = blockIdx.x;
  const int pass = task & 1;        // 0 = negative weights, 1 = positive
  const int bw   = task >> 1;
  const int bat  = bw >> 6;
  const int win  = bw & 63;
  const int wr   = win >> 3, wc = win & 7;
  const int tid  = (int)threadIdx.x;

  const size_t off = (size_t)bat * (256 * 256) + (size_t)wr * (WINSZ * 256)
                   + (size_t)wc * WINSZ;

  // ---- stage the two strided 32x32 f32 tiles into LDS via the TDM ----
#if __has_builtin(__builtin_amdgcn_tensor_load_to_lds) && __has_builtin(__builtin_amdgcn_s_wait_tensorcnt)
  {
    const uint32_t td = 1u << 20;   // tensor_dim0/1: large -> no OOB clamp
    v8i g1;
    g1[0] = (int)(2u << 16);                                      // data_size = 4B
    g1[1] = (int)((td & 0xFFFFu) << 16);                          // tensor_dim0 lo
    g1[2] = (int)(((td >> 16) & 0xFFFFu) | ((td & 0xFFFFu) << 16)); // dim0 hi | dim1 lo
    g1[3] = (int)(((td >> 16) & 0xFFFFu) | ((uint32_t)WINSZ << 16)); // dim1 hi | tile_dim0
    g1[4] = WINSZ;                                                // tile_dim1=32, tile_dim2=0
    g1[5] = 256;                                                  // tensor_dim0_stride (elems)
    g1[6] = 0;
    g1[7] = 0;
    v4i gz = {0, 0, 0, 0};

    uint64_t gp = (uint64_t)(uintptr_t)(y_pred + off);
    uint64_t gt = (uint64_t)(uintptr_t)(y_true + off);
    // Generic LDS pointer: addr[31:0] is the LDS byte offset (aperture form).
    uint32_t lp = (uint32_t)(uintptr_t)(&tile_p[0]);
    uint32_t lt = (uint32_t)(uintptr_t)(&tile_t[0]);

    v4u g0p = { 1u, lp, (uint32_t)gp,
                (uint32_t)((gp >> 32) & 0x01FFFFFFull) | (2u << 30) }; // count=1, type=2
    v4u g0t = { 1u, lt, (uint32_t)gt,
                (uint32_t)((gt >> 32) & 0x01FFFFFFull) | (2u << 30) };

#if defined(__clang_major__) && (__clang_major__ >= 23)
    v8i gz8 = {0, 0, 0, 0, 0, 0, 0, 0};
    __builtin_amdgcn_tensor_load_to_lds(g0p, g1, gz, gz, gz8, 0);
    __builtin_amdgcn_tensor_load_to_lds(g0t, g1, gz, gz, gz8, 0);
#else
    __builtin_amdgcn_tensor_load_to_lds(g0p, g1, gz, gz, 0);
    __builtin_amdgcn_tensor_load_to_lds(g0t, g1, gz, gz, 0);
#endif
    __builtin_amdgcn_s_wait_tensorcnt(0);
  }
#else
  for (int i = tid; i < NPIX; i += 32) {
    int r = i >> 5, c = i & 31;
    tile_p[i] = y_pred[off + (size_t)r * 256 + c];
    tile_t[i] = y_true[off + (size_t)r * 256 + c];
  }
#endif
  __syncthreads();

  // ---- edge costs for this pass (all 32 lanes) ----
  for (int e = tid; e < NEDGE; e += 32) {
    int a, b2; edge_nodes(e, a, b2);
    float gtv = tile_t[a] + tile_t[b2];
    float cv  = tile_p[a] + tile_p[b2];
    if (pass == 0) { if (gtv > 20.0f) cv = 20.0f; }  // costs_n[gt>20] = 20
    else           { if (gtv < 10.0f) cv = 0.0f;  }  // costs_p[gt<10] = 0
    cost[e]  = cv;
    order[e] = (unsigned short)e;
  }
  __syncthreads();

  if (tid == 0) {
    // ---- stable descending merge sort of edge indices (argsort(-costs)) ----
    unsigned short* src = order;
    unsigned short* dst = auxb;
    for (int wdt = 1; wdt < NEDGE; wdt <<= 1) {
      for (int lo = 0; lo < NEDGE; lo += (wdt << 1)) {
        int mid = lo + wdt;        if (mid > NEDGE) mid = NEDGE;
        int hi  = lo + (wdt << 1); if (hi  > NEDGE) hi  = NEDGE;
        int i = lo, j = mid, k = lo;
        while (i < mid && j < hi) {
          unsigned short ei = src[i], ej = src[j];
          if (cost[ei] >= cost[ej]) { dst[k++] = ei; ++i; }  // ties: left first -> stable
          else                      { dst[k++] = ej; ++j; }
        }
        while (i < mid) dst[k++] = src[i++];
        while (j < hi)  dst[k++] = src[j++];
      }
      unsigned short* t = src; src = dst; dst = t;
    }
    unsigned short* sorted = src;

    // ---- background (t==0) connected components, 4-neighborhood ----
    for (int i = 0; i < NPIX; ++i) parent[i] = (unsigned short)i;
    for (int e = 0; e < NEDGE; ++e) {
      int a, b2; edge_nodes(e, a, b2);
      if (tile_t[a] < 10.0f && tile_t[b2] < 10.0f) {
        int ra = uf_find(parent, a), rb = uf_find(parent, b2);
        if (ra != rb) parent[ra] = (unsigned short)rb;
      }
    }
    for (int i = 0; i < NPIX; ++i)
      seg[i] = (tile_t[i] < 10.0f) ? (unsigned short)(uf_find(parent, i) + 1)
                                   : (unsigned short)0;

    // ---- MALIS Kruskal with sorted intrusive-list label histograms ----
    for (int i = 0; i < NPIX; ++i) {
      parent[i] = (unsigned short)i;
      if (seg[i]) {
        hlab[i] = seg[i]; hcnt[i] = 1; hnext[i] = -1;
        head[i] = (short)i; csize[i] = 1;
      } else {
        head[i] = -1; csize[i] = 0;
      }
    }

    double lossAcc = 0.0;
    unsigned long long sumW = 0;

    for (int ei = 0; ei < NEDGE; ++ei) {
      int e = (int)sorted[ei];
      int a, b2; edge_nodes(e, a, b2);
      int ra = uf_find(parent, a), rb = uf_find(parent, b2);
      if (ra == rb) continue;

      // merge sorted histograms; common = sum_l ca[l]*cb[l]
      unsigned long long common = 0;
      int ia = (int)head[ra], ib = (int)head[rb];
      int mh = -1, mt = -1;
      while (ia >= 0 || ib >= 0) {
        if (ia >= 0 && ib >= 0 && hlab[ia] == hlab[ib]) {
          common += (unsigned long long)hcnt[ia] * (unsigned long long)hcnt[ib];
          hcnt[ib] = (unsigned short)(hcnt[ib] + hcnt[ia]);
          ia = (int)hnext[ia];          // fold ia's entry into ib's
          continue;
        }
        int take;
        if (ib < 0 || (ia >= 0 && hlab[ia] < hlab[ib])) { take = ia; ia = (int)hnext[ia]; }
        else                                            { take = ib; ib = (int)hnext[ib]; }
        if (mt < 0) mh = take; else hnext[mt] = (short)take;
        mt = take;
      }
      if (mt >= 0) hnext[mt] = -1;

      unsigned long long w = pass
          ? common
          : (unsigned long long)csize[ra] * (unsigned long long)csize[rb] - common;
      sumW += w;

      if (w) {
        // mask (applied after normalization in the reference; normalization is
        // the final division by sumW, so masking here is equivalent)
        float gtv  = tile_t[a] + tile_t[b2];
        bool  keep = pass ? (gtv >= 20.0f) : (gtv < 10.0f);
        if (keep) {
          float pa = tile_p[a], pb = tile_p[b2];
          float term = pass
              ? ((20.0f - pa) * (20.0f - pa) + (20.0f - pb) * (20.0f - pb))
              : (pa * pa + pb * pb);
          lossAcc += (double)w * (double)term;
        }
      }

      parent[ra] = (unsigned short)rb;
      head[rb]   = (short)mh;
      head[ra]   = -1;
      csize[rb]  = (unsigned short)(csize[rb] + csize[ra]);
    }

    partials[task] = (sumW > 0) ? (float)(lossAcc / (double)sumW) : 0.0f;
  }
}

// Deterministic fixed-order reduction of the 512 per-task partials.
__global__ void malis_reduce_kernel(const float* __restrict__ partials,
                                    float* __restrict__ out)
{
  if (blockIdx.x == 0 && threadIdx.x == 0) {
    double s = 0.0;
    for (int i = 0; i < NTASK; ++i) s += (double)partials[i];
    out[0] = (float)s;
  }
}

extern "C" void kernel_launch(void* const* d_in, const int* in_sizes, int n_in,
                              void* d_out, int out_size, void* d_ws, size_t ws_size,
                              hipStream_t stream) {
  (void)in_sizes; (void)n_in; (void)out_size; (void)ws_size;
  const float* y_true = (const float*)d_in[0];
  const float* y_pred = (const float*)d_in[1];
  float* out      = (float*)d_out;
  float* partials = (float*)d_ws;   // 512 floats of scratch

  malis_window_kernel<<<NTASK, 32, 0, stream>>>(y_true, y_pred, partials);
  malis_reduce_kernel<<<1, 32, 0, stream>>>(partials, out);
}